// ECELoss_52913997087021
// MI455X (gfx1250) — compile-verified
//
#include <hip/hip_runtime.h>
#include <math.h>

// ---------------- CDNA5 async global->LDS plumbing ----------------
#ifndef __has_builtin
#define __has_builtin(x) 0
#endif

typedef int v4i __attribute__((__vector_size__(16)));
typedef __attribute__((address_space(1))) v4i* gvec_p; // global int4*
typedef __attribute__((address_space(3))) v4i* lvec_p; // LDS int4*

#if __has_builtin(__builtin_amdgcn_global_load_async_to_lds_b128)
#define ECE_ASYNC_LDS 1
#else
#define ECE_ASYNC_LDS 0
#endif

template <int CNT>
__device__ __forceinline__ void wait_asynccnt() {
#if __has_builtin(__builtin_amdgcn_s_wait_asynccnt)
  __builtin_amdgcn_s_wait_asynccnt(CNT);
#else
  asm volatile("s_wait_asynccnt %0" ::"i"(CNT) : "memory");
#endif
}

// ---------------- problem constants ----------------
constexpr int NBINS = 20;
constexpr int C = 128;        // classes per row (reference fixes C=128)
constexpr int ROWS = 32;      // rows per LDS stage  (32*128*4B = 16 KB)
constexpr int TPB = 256;      // 8 wave32s
constexpr int CHUNKS = (ROWS * C * 4) / (TPB * 16); // 16B chunks per thread = 4

// ---------------- per-row softmax-confidence / argmax ----------------
__device__ __forceinline__ void process_row(const float* __restrict__ rowdata,
                                            int row,
                                            const int* __restrict__ labels,
                                            float* h_cnt, float* h_conf,
                                            float* h_acc) {
  const int lane = threadIdx.x & 31;
  float4 v = *(const float4*)(rowdata + lane * 4);

  // lane-local max + first-index tiebreak
  float m = v.x;
  int mi = lane * 4;
  if (v.y > m) { m = v.y; mi = lane * 4 + 1; }
  if (v.z > m) { m = v.z; mi = lane * 4 + 2; }
  if (v.w > m) { m = v.w; mi = lane * 4 + 3; }
#pragma unroll
  for (int off = 16; off >= 1; off >>= 1) {
    float om = __shfl_xor(m, off, 32);
    int omi = __shfl_xor(mi, off, 32);
    if (om > m || (om == m && omi < mi)) { m = om; mi = omi; }
  }

  // sum of exp(l - max) ; TEMPERATURE == 1
  float e = expf(v.x - m) + expf(v.y - m) + expf(v.z - m) + expf(v.w - m);
#pragma unroll
  for (int off = 16; off >= 1; off >>= 1) e += __shfl_xor(e, off, 32);

  if (lane == 0) {
    float conf = 1.0f / e;                 // max prob = exp(0)/sum
    int label = labels[row];
    float acc = (mi == label) ? 1.0f : 0.0f;
    // bin i covers (i/20, (i+1)/20]  ->  ceil(conf*20)-1, clamped
    int b = (int)ceilf(conf * 20.0f) - 1;
    b = b < 0 ? 0 : (b > NBINS - 1 ? NBINS - 1 : b);
    atomicAdd(&h_cnt[b], 1.0f);
    atomicAdd(&h_conf[b], conf);
    atomicAdd(&h_acc[b], acc);
  }
}

#if ECE_ASYNC_LDS
__device__ __forceinline__ void issue_stage(const float* __restrict__ logits,
                                            int stage, float* lds_dst,
                                            int tid) {
  const size_t gbase = (size_t)stage * (ROWS * C * 4); // bytes
  const char* gp = (const char*)logits;
  char* lp = (char*)lds_dst;
#pragma unroll
  for (int k = 0; k < CHUNKS; ++k) {
    const int c = tid + k * TPB;
    __builtin_amdgcn_global_load_async_to_lds_b128(
        (gvec_p)(gp + gbase + (size_t)c * 16), (lvec_p)(lp + c * 16),
        /*imm offset*/ 0, /*cpol*/ 0);
  }
}
#endif

// ---------------- kernels ----------------
__global__ void ece_zero_ws(float* __restrict__ ws) {
  if (threadIdx.x < 3 * NBINS) ws[threadIdx.x] = 0.0f;
}

__global__ void __launch_bounds__(TPB) ece_main(
    const float* __restrict__ logits, const int* __restrict__ labels,
    float* __restrict__ ws, int N) {
  __shared__ float h_cnt[NBINS], h_conf[NBINS], h_acc[NBINS];
  const int tid = threadIdx.x;
  const int wave = tid >> 5;
  if (tid < NBINS) { h_cnt[tid] = 0.f; h_conf[tid] = 0.f; h_acc[tid] = 0.f; }
  __syncthreads();

  const int fullStages = N / ROWS;

#if ECE_ASYNC_LDS
  __shared__ alignas(16) float tile[2][ROWS * C];
  int s = (int)blockIdx.x;
  int buf = 0;
  if (s < fullStages) issue_stage(logits, s, &tile[0][0], tid);
  for (; s < fullStages; s += (int)gridDim.x) {
    const int snext = s + (int)gridDim.x;
    if (snext < fullStages) {
      issue_stage(logits, snext, &tile[buf ^ 1][0], tid);
      wait_asynccnt<CHUNKS>();   // only next-stage's CHUNKS may remain pending
    } else {
      wait_asynccnt<0>();
    }
    __syncthreads();             // stage visible to all waves
    const int base = s * ROWS;
#pragma unroll
    for (int j = 0; j < ROWS / (TPB / 32); ++j) {
      const int r = wave * (ROWS / (TPB / 32)) + j;
      process_row(&tile[buf][r * C], base + r, labels, h_cnt, h_conf, h_acc);
    }
    __syncthreads();             // done reading before buffer is refilled
    buf ^= 1;
  }
#else
  // fallback: direct coalesced global_load_b128, wave-per-row grid-stride
  const int gwave = (int)blockIdx.x * (TPB / 32) + wave;
  const int nwaves = (int)gridDim.x * (TPB / 32);
  for (int row = gwave; row < fullStages * ROWS; row += nwaves)
    process_row(logits + (size_t)row * C, row, labels, h_cnt, h_conf, h_acc);
#endif

  // tail rows (N % ROWS): block 0, wave-per-row, direct loads
  if (blockIdx.x == 0) {
    for (int row = fullStages * ROWS + wave; row < N; row += TPB / 32)
      process_row(logits + (size_t)row * C, row, labels, h_cnt, h_conf, h_acc);
  }

  __syncthreads();
  if (tid < NBINS) {
    atomicAdd(&ws[tid], h_cnt[tid]);
    atomicAdd(&ws[NBINS + tid], h_conf[tid]);
    atomicAdd(&ws[2 * NBINS + tid], h_acc[tid]);
  }
}

__global__ void ece_finalize(const float* __restrict__ ws,
                             float* __restrict__ out, float invN) {
  const int t = threadIdx.x;
  float cnt = 0.f, cs = 0.f, as = 0.f;
  if (t < NBINS) { cnt = ws[t]; cs = ws[NBINS + t]; as = ws[2 * NBINS + t]; }
  float contrib = 0.f, ys = 0.f;
  if (t < NBINS && cnt > 0.f) {
    const float avg_conf = cs / cnt;
    const float avg_acc = as / cnt;
    ys = avg_acc;
    contrib = fabsf(avg_conf - avg_acc) * cnt * invN;
  }
  if (t < NBINS) out[1 + t] = ys;
#pragma unroll
  for (int off = 16; off >= 1; off >>= 1) contrib += __shfl_xor(contrib, off, 32);
  if (t == 0) out[0] = contrib;
}

// ---------------- launcher ----------------
extern "C" void kernel_launch(void* const* d_in, const int* in_sizes, int n_in,
                              void* d_out, int out_size, void* d_ws,
                              size_t ws_size, hipStream_t stream) {
  const float* logits = (const float*)d_in[0];
  const int* labels = (const int*)d_in[1];
  float* out = (float*)d_out;
  float* ws = (float*)d_ws;
  const int N = in_sizes[1]; // labels count == rows

  ece_zero_ws<<<1, 64, 0, stream>>>(ws);

  const int fullStages = N / ROWS;
  int grid = fullStages > 0 ? fullStages : 1;
  if (grid > 3072) grid = 3072; // persistent blocks; enough to fill all WGPs
  ece_main<<<grid, TPB, 0, stream>>>(logits, labels, ws, N);

  ece_finalize<<<1, 32, 0, stream>>>(ws, out, 1.0f / (float)N);
}